// VectorQuantiser_41901700940120
// MI455X (gfx1250) — compile-verified
//
#include <hip/hip_runtime.h>

// VQ-VAE vector quantizer, fused for MI455X (gfx1250, wave32).
// dot GEMM via V_WMMA_F32_16X16X4_F32; epilogue fused (dist/similarity/argmin/
// gather/loss). Store-bound on similarity (134 MB) -> keep everything f32.

typedef __attribute__((ext_vector_type(2))) float v2f;
typedef __attribute__((ext_vector_type(8))) float v8f;

#define VQ_D     256
#define VQ_K     1024
#define VQ_NTOK  32768
#define TILE_M   16
#define LDA      260            // LDS row stride (mod 64 == 4 -> conflict-free b64 reads)
#define VQ_BETA  0.25f

__global__ __launch_bounds__(256)
void vq_norms_kernel(const float* __restrict__ cb,
                     float* __restrict__ ne,
                     float* __restrict__ loss_slot) {
    int k = blockIdx.x * 256 + threadIdx.x;
    if (k < VQ_K) {
        const float4* row = (const float4*)(cb + (size_t)k * VQ_D);
        float s = 0.f;
#pragma unroll 8
        for (int j = 0; j < VQ_D / 4; ++j) {
            float4 v = row[j];
            s += v.x * v.x + v.y * v.y + v.z * v.z + v.w * v.w;
        }
        ne[k] = s;
    }
    if (blockIdx.x == 0 && threadIdx.x == 0) *loss_slot = 0.f;  // graph-replay safe re-zero
}

__global__ __launch_bounds__(256)
void vq_main_kernel(const float* __restrict__ z,
                    const float* __restrict__ cb,
                    const float* __restrict__ ne,
                    float* __restrict__ out_zq,
                    float* __restrict__ out_sim,
                    float* __restrict__ out_ids,
                    float* __restrict__ out_loss) {
    __shared__ float ldsA[TILE_M * LDA];     // z_e tile, padded rows
    __shared__ float ldsPart[256];
    __shared__ float ldsNz[TILE_M];
    __shared__ float ldsBestD[8 * TILE_M];
    __shared__ int   ldsBestI[8 * TILE_M];
    __shared__ int   ldsIds[TILE_M];
    __shared__ float ldsTok[TILE_M];

    const int t    = threadIdx.x;
    const int wave = t >> 5;
    const int lane = t & 31;
    const int half = lane >> 4;   // 0 or 1
    const int ln   = lane & 15;
    const long tok0 = (long)blockIdx.x * TILE_M;

    // ---- stage z_e tile (16 x 256) into LDS, coalesced, + squared partials ----
    {
        const int m  = t >> 4;          // token row 0..15
        const int c0 = (t & 15) * 16;   // 16 floats per thread
        const float4* src = (const float4*)(z + (tok0 + m) * VQ_D + c0);
        float* dst = &ldsA[m * LDA + c0];
        float s = 0.f;
#pragma unroll
        for (int j = 0; j < 4; ++j) {
            float4 v = src[j];
            *(float4*)(dst + 4 * j) = v;
            s += v.x * v.x + v.y * v.y + v.z * v.z + v.w * v.w;
        }
        ldsPart[t] = s;
    }
    __syncthreads();
    if (t < TILE_M) {
        float s = 0.f;
#pragma unroll
        for (int j = 0; j < 16; ++j) s += ldsPart[t * 16 + j];
        ldsNz[t] = s;
    }
    __syncthreads();

    // per-lane token norms for the C/D rows this lane touches (M = g + 8*half)
    float nzv[8], rnzv[8];
#pragma unroll
    for (int g = 0; g < 8; ++g) {
        float v = ldsNz[g + 8 * half];
        nzv[g]  = v;
        rnzv[g] = rsqrtf(v);
    }

    float bd[8];
    int   bi[8];
#pragma unroll
    for (int g = 0; g < 8; ++g) { bd[g] = 3.4e38f; bi[g] = 0; }

    const int koff = 2 * half;                 // K sub-offset per A/B fragment layout
    const float* arow = &ldsA[ln * LDA];       // A row m = ln (lanes 0-15 / 16-31 mirror)

    for (int iter = 0; iter < 8; ++iter) {
        const int code_base = iter * 128 + wave * 16;
        const int code      = code_base + ln;           // this lane's B column
        const float  ne_l   = ne[code];
        const float  rne_l  = rsqrtf(ne_l);
        const float* brow   = cb + (long)code * VQ_D;

        v8f acc = {0.f, 0.f, 0.f, 0.f, 0.f, 0.f, 0.f, 0.f};
#pragma unroll
        for (int kk = 0; kk < 64; ++kk) {
            const int k0 = kk * 4 + koff;
            v2f a = *(const v2f*)(arow + k0);   // ds_load_b64, bank-conflict-free
            v2f b = *(const v2f*)(brow + k0);   // global b64, codebook L2-resident
            acc = __builtin_amdgcn_wmma_f32_16x16x4_f32(
                /*neg_a=*/false, a, /*neg_b=*/false, b,
                /*c_mod=*/(short)0, acc, /*reuse_a=*/false, /*reuse_b=*/false);
        }

        // fused epilogue: dist, similarity store, running argmin
#pragma unroll
        for (int g = 0; g < 8; ++g) {
            const float dot  = acc[g];
            const int   row  = g + 8 * half;                   // token within tile
            const float dist = fmaf(-2.f, dot, nzv[g] + ne_l);
            out_sim[(tok0 + row) * (long)VQ_K + code] = dot * rnzv[g] * rne_l;
            if (dist < bd[g] || (dist == bd[g] && code < bi[g])) {
                bd[g] = dist; bi[g] = code;
            }
        }
    }

    // argmin across the 16 lanes of each half (wave32 butterfly, halves stay separate)
#pragma unroll
    for (int g = 0; g < 8; ++g) {
#pragma unroll
        for (int m = 8; m >= 1; m >>= 1) {
            float od = __shfl_xor(bd[g], m, 32);
            int   oi = __shfl_xor(bi[g], m, 32);
            if (od < bd[g] || (od == bd[g] && oi < bi[g])) { bd[g] = od; bi[g] = oi; }
        }
    }
    if (ln == 0) {   // lane 0 -> rows 0..7, lane 16 -> rows 8..15
#pragma unroll
        for (int g = 0; g < 8; ++g) {
            ldsBestD[wave * TILE_M + g + 8 * half] = bd[g];
            ldsBestI[wave * TILE_M + g + 8 * half] = bi[g];
        }
    }
    __syncthreads();

    // cross-wave reduction -> final ids, per-token loss term
    if (t < TILE_M) {
        float d = ldsBestD[t];
        int   i = ldsBestI[t];
#pragma unroll
        for (int w = 1; w < 8; ++w) {
            float od = ldsBestD[w * TILE_M + t];
            int   oi = ldsBestI[w * TILE_M + t];
            if (od < d || (od == d && oi < i)) { d = od; i = oi; }
        }
        ldsIds[t] = i;
        ldsTok[t] = sqrtf(fmaxf(d, 0.f));   // ||z_e - z_q|| for this token
        out_ids[tok0 + t] = (float)i;
    }
    __syncthreads();

    // gather z_q = codebook[ids], coalesced
    {
        const int m  = t >> 4;
        const int c0 = (t & 15) * 16;
        const int id = ldsIds[m];
        const float4* src = (const float4*)(cb + (long)id * VQ_D + c0);
        float4* dst = (float4*)(out_zq + (tok0 + m) * VQ_D + c0);
#pragma unroll
        for (int j = 0; j < 4; ++j) dst[j] = src[j];
    }
    if (t == 0) {
        float s = 0.f;
#pragma unroll
        for (int j = 0; j < TILE_M; ++j) s += ldsTok[j];
        // loss_vq = (1 + BETA) * mean ||z_e - z_q||
        atomicAdd(out_loss, s * ((1.f + VQ_BETA) / (float)VQ_NTOK));
    }
}

extern "C" void kernel_launch(void* const* d_in, const int* in_sizes, int n_in,
                              void* d_out, int out_size, void* d_ws, size_t ws_size,
                              hipStream_t stream) {
    (void)in_sizes; (void)n_in; (void)out_size; (void)ws_size;
    const float* z  = (const float*)d_in[0];   // z_e   (8,4096,256) f32
    const float* cb = (const float*)d_in[1];   // codebook (1024,256) f32

    float* out      = (float*)d_out;
    float* out_zq   = out;                                   //  8*4096*256
    float* out_sim  = out_zq + (size_t)8 * 4096 * 256;       //  8*4096*1024
    float* out_ids  = out_sim + (size_t)8 * 4096 * 1024;     //  8*4096
    float* out_loss = out_ids + (size_t)8 * 4096;            //  1

    float* ws_ne = (float*)d_ws;                             // 1024 codebook norms

    vq_norms_kernel<<<VQ_K / 256, 256, 0, stream>>>(cb, ws_ne, out_loss);
    vq_main_kernel<<<VQ_NTOK / TILE_M, 256, 0, stream>>>(
        z, cb, ws_ne, out_zq, out_sim, out_ids, out_loss);
}